// PatchTSTPolynomialEmbedding_70411693851190
// MI455X (gfx1250) — compile-verified
//
#include <hip/hip_runtime.h>
#include <stdint.h>

// PatchTSTPolynomialEmbedding, algebraically folded for MI455X (gfx1250).
// out[b,c,p,:] = (Pw_x + U*G*M^T) x_c + (proj_b + Z*G*kq0) + Pw_rff * rff_c
// G = X^T X per (b,p);  U = Pw1*Wv1, Z = Pw0*Wv0, M = Wq1^T*Wk1 precomputed.

typedef float v2f __attribute__((ext_vector_type(2)));
typedef float v8f __attribute__((ext_vector_type(8)));
typedef unsigned int v4u __attribute__((ext_vector_type(4)));
typedef int v4i __attribute__((ext_vector_type(4)));
typedef int v8i __attribute__((ext_vector_type(8)));

#define NB   16
#define NC   64
#define NP   128
#define NL   16
#define ND   256
#define FEAT 656

#if defined(__has_builtin)
#if __has_builtin(__builtin_amdgcn_tensor_load_to_lds)
#define HAVE_TDM 1
#endif
#endif
#ifndef HAVE_TDM
#define HAVE_TDM 0
#endif

static __device__ __forceinline__ v8f wmma4(v2f a, v2f b, v8f c) {
    return __builtin_amdgcn_wmma_f32_16x16x4_f32(
        false, a, false, b, (short)0, c, false, false);
}

// ---------------- precompute: kq0 (16), M (16x16), U (256x16), Z (256x16) ----
__global__ __launch_bounds__(256) void precompute_kernel(
    const float* __restrict__ Wq,   // (2,256,16)
    const float* __restrict__ Wk,   // (2,256,16)
    const float* __restrict__ Wv,   // (2,256,16)
    const float* __restrict__ projw,// (256,656)
    float* __restrict__ ws)
{
    __shared__ float q0s[256];
    const int t = threadIdx.x;

    // q0[d] = sum_l Wq[0][d][l]
    {
        float s = 0.f;
        #pragma unroll
        for (int l = 0; l < NL; ++l) s += Wq[t * NL + l];
        q0s[t] = s;
    }
    __syncthreads();

    // kq0[l] = sum_d Wk[0][d][l] * q0[d]
    if (t < NL) {
        float kq = 0.f;
        for (int d = 0; d < ND; ++d) kq += Wk[d * NL + t] * q0s[d];
        ws[t] = kq;
    }

    // M[l][m] = sum_d Wq[1][d][l] * Wk[1][d][m]
    {
        const int l = t >> 4, m = t & 15;
        float mm = 0.f;
        for (int d = 0; d < ND; ++d)
            mm += Wq[ND * NL + d * NL + l] * Wk[ND * NL + d * NL + m];
        ws[16 + l * 16 + m] = mm;
    }

    // U[d][l] = sum_f projw[d][272+f] * Wv[1][f][l]
    // Z[d][l] = sum_f projw[d][ 16+f] * Wv[0][f][l]
    for (int l = 0; l < NL; ++l) {
        float u = 0.f, z = 0.f;
        for (int f = 0; f < 256; ++f) {
            const float p1 = projw[t * FEAT + 272 + f];
            const float p0 = projw[t * FEAT + 16 + f];
            u += p1 * Wv[256 * NL + f * NL + l];
            z += p0 * Wv[f * NL + l];
        }
        ws[272 + t * NL + l]  = u;
        ws[4368 + t * NL + l] = z;
    }
}

// ---------------- fused per-(b,p) kernel ------------------------------------
__global__ __launch_bounds__(256) void fused_kernel(
    const float* __restrict__ x,      // (16,64,128,16)
    const float* __restrict__ freqw,  // (16,64)
    const float* __restrict__ freqb,  // (64)
    const float* __restrict__ projw,  // (256,656)
    const float* __restrict__ projb,  // (256)
    const float* __restrict__ ws,
    float* __restrict__ out)          // (16,64,128,256)
{
    __shared__ float Xs[NC * 17];       // 64x16 channel tile, padded stride 17
    __shared__ float Gs[256];           // Gram 16x16
    __shared__ float Hs[256];           // G * M^T
    __shared__ float gks[16];           // G * kq0
    __shared__ float biasS[ND];         // proj_b + Z*(G*kq0)
    __shared__ float Weff[ND * 16];     // Pw_x + U*H  (row-major d x 16)
    __shared__ float rffs[NC * 129];    // sin|cos features, padded stride

    const int bp   = blockIdx.x;
    const int b    = bp >> 7;           // / 128
    const int p    = bp & 127;
    const int t    = threadIdx.x;
    const int wave = t >> 5;
    const int lane = t & 31;
    const int n    = lane & 15;         // N column / A row index
    const int hi   = lane >> 4;         // which K-pair this half-wave carries

    const float* kq0 = ws;
    const float* Mm  = ws + 16;
    const float* U   = ws + 272;
    const float* Z   = ws + 4368;

    // 1. stage X = x[b, :, p, :] (64 rows x 16 f32, row stride 8KB) into LDS
#if HAVE_TDM
    if (wave == 0) {
        const uint64_t ga =
            (uint64_t)(uintptr_t)(x + (size_t)((b * NC) * NP + p) * NL);
        const uint32_t lds = (uint32_t)(uintptr_t)(&Xs[0]);
        v4u g0;
        g0.x = 1u;                                   // count=1 valid descriptor
        g0.y = lds;                                  // lds_addr (bytes)
        g0.z = (uint32_t)ga;                         // global_addr[31:0]
        g0.w = (uint32_t)((ga >> 32) & 0x01FFFFFFu)  // global_addr[56:32]
               | (2u << 30);                         // type=2 (image)
        v8i g1;
        g1[0] = (int)((2u << 16)    // data_size = 4 bytes
                    | (1u << 20)    // pad_enable
                    | (3u << 22)    // pad_interval: 16 DWORDs
                    | (0u << 25));  // pad_amount: 1 DWORD -> LDS stride 17
        g1[1] = (int)(16u << 16);   // tensor_dim0 = 16 elements
        g1[2] = (int)(64u << 16);   // tensor_dim1 = 64 rows
        g1[3] = (int)(16u << 16);   // tile_dim0 = 16
        g1[4] = (int)(64u);         // tile_dim1 = 64, tile_dim2 = 0
        g1[5] = (int)(2048u);       // tensor_dim0_stride = P*L = 2048 elements
        g1[6] = 0;
        g1[7] = 0;
        v4i z4 = {0, 0, 0, 0};
        v8i z8 = {0, 0, 0, 0, 0, 0, 0, 0};
        __builtin_amdgcn_tensor_load_to_lds(g0, g1, z4, z4, z8, 0);
        __builtin_amdgcn_s_wait_tensorcnt(0);
    }
#else
    {
        const int c = t >> 2, q = t & 3;
        const float4 v = *reinterpret_cast<const float4*>(
            x + (((size_t)(b * NC + c) * NP + p) * NL + q * 4));
        float* dst = &Xs[c * 17 + q * 4];
        dst[0] = v.x; dst[1] = v.y; dst[2] = v.z; dst[3] = v.w;
    }
#endif
    __syncthreads();

    // 2. G = X^T X (each thread one of 256 entries, 64-deep dot)
    {
        const int l = t >> 4, m = t & 15;
        float g = 0.f;
        for (int c = 0; c < NC; ++c) g += Xs[c * 17 + l] * Xs[c * 17 + m];
        Gs[l * 16 + m] = g;
    }
    __syncthreads();

    // 3. H = G*M^T ; gk = G*kq0
    {
        const int l = t >> 4, m = t & 15;
        float h = 0.f;
        #pragma unroll
        for (int k = 0; k < 16; ++k) h += Gs[l * 16 + k] * Mm[m * 16 + k];
        Hs[l * 16 + m] = h;
        if (t < 16) {
            float gk = 0.f;
            #pragma unroll
            for (int k = 0; k < 16; ++k) gk += Gs[t * 16 + k] * kq0[k];
            gks[t] = gk;
        }
    }
    __syncthreads();

    // 4. bias_eff[d] = proj_b[d] + Z[d,:] . gk
    {
        float be = projb[t];
        #pragma unroll
        for (int l = 0; l < NL; ++l) be += Z[t * NL + l] * gks[l];
        biasS[t] = be;
    }

    // 7. RFF features: wx = X*freq_w + freq_b ; rff = [sin(wx) | cos(wx)]
    {
        const int c = t >> 2, j0 = (t & 3) * 16;
        #pragma unroll
        for (int jj = 0; jj < 16; ++jj) {
            const int j = j0 + jj;
            float wx = freqb[j];
            #pragma unroll
            for (int l = 0; l < NL; ++l) wx += Xs[c * 17 + l] * freqw[l * 64 + j];
            rffs[c * 129 + j]      = __sinf(wx);
            rffs[c * 129 + 64 + j] = __cosf(wx);
        }
    }

    // 5. Weff = Pw_x + U*H via v_wmma_f32_16x16x4 (2 d-tiles per wave)
    for (int tt = 0; tt < 2; ++tt) {
        const int d0 = (wave * 2 + tt) * 16;
        v8f acc;
        #pragma unroll
        for (int r = 0; r < 8; ++r)
            acc[r] = projw[(size_t)(d0 + r + hi * 8) * FEAT + n];   // Pw_x tile
        #pragma unroll
        for (int kc = 0; kc < 16; kc += 4) {
            v2f a, bb;
            a.x  = U[(d0 + n) * 16 + kc + 2 * hi];
            a.y  = U[(d0 + n) * 16 + kc + 2 * hi + 1];
            bb.x = Hs[(kc + 2 * hi) * 16 + n];
            bb.y = Hs[(kc + 2 * hi + 1) * 16 + n];
            acc = wmma4(a, bb, acc);
        }
        #pragma unroll
        for (int r = 0; r < 8; ++r)
            Weff[(d0 + r + hi * 8) * 16 + n] = acc[r];
    }
    __syncthreads();

    // 6+8. Each wave owns 2 d-tiles; register-cache the B panels (Weff + Pw_rff)
    //      once per d-tile, then sweep the 4 c-tiles:
    //      acc = bias + X*Weff^T (K=16) + rff*Pw_rff^T (K=128)
    for (int half = 0; half < 2; ++half) {
        const int d0 = (wave * 2 + half) * 16;
        const float bv = biasS[d0 + n];

        v2f Bw[4];
        #pragma unroll
        for (int k4 = 0; k4 < 4; ++k4) {
            Bw[k4].x = Weff[(d0 + n) * 16 + k4 * 4 + 2 * hi];
            Bw[k4].y = Weff[(d0 + n) * 16 + k4 * 4 + 2 * hi + 1];
        }
        const float* pr = projw + (size_t)(d0 + n) * FEAT + 528;  // Pw_rff row
        v2f Brf[32];
        #pragma unroll
        for (int k4 = 0; k4 < 32; ++k4) {
            Brf[k4].x = pr[k4 * 4 + 2 * hi];
            Brf[k4].y = pr[k4 * 4 + 2 * hi + 1];
        }

        for (int mt = 0; mt < 4; ++mt) {
            const int c0 = mt * 16;
            v8f acc;
            #pragma unroll
            for (int r = 0; r < 8; ++r) acc[r] = bv;

            #pragma unroll
            for (int k4 = 0; k4 < 4; ++k4) {
                v2f a;
                a.x = Xs[(c0 + n) * 17 + k4 * 4 + 2 * hi];
                a.y = Xs[(c0 + n) * 17 + k4 * 4 + 2 * hi + 1];
                acc = wmma4(a, Bw[k4], acc);
            }
            #pragma unroll
            for (int k4 = 0; k4 < 32; ++k4) {
                v2f a;
                a.x = rffs[(c0 + n) * 129 + k4 * 4 + 2 * hi];
                a.y = rffs[(c0 + n) * 129 + k4 * 4 + 2 * hi + 1];
                acc = wmma4(a, Brf[k4], acc);
            }

            #pragma unroll
            for (int r = 0; r < 8; ++r) {
                const int c = c0 + r + hi * 8;
                out[((size_t)(b * NC + c) * NP + p) * ND + d0 + n] = acc[r];
            }
        }
    }
}

extern "C" void kernel_launch(void* const* d_in, const int* in_sizes, int n_in,
                              void* d_out, int out_size, void* d_ws, size_t ws_size,
                              hipStream_t stream) {
    const float* x     = (const float*)d_in[0];
    const float* Wq    = (const float*)d_in[1];
    const float* Wk    = (const float*)d_in[2];
    const float* Wv    = (const float*)d_in[3];
    const float* freqw = (const float*)d_in[4];
    const float* freqb = (const float*)d_in[5];
    const float* projw = (const float*)d_in[6];
    const float* projb = (const float*)d_in[7];
    float* out = (float*)d_out;
    float* ws  = (float*)d_ws;

    precompute_kernel<<<1, 256, 0, stream>>>(Wq, Wk, Wv, projw, ws);
    fused_kernel<<<NB * NP, 256, 0, stream>>>(x, freqw, freqb, projw, projb, ws, out);
}